// ComplexModel_56942676411136
// MI455X (gfx1250) — compile-verified
//
#include <hip/hip_runtime.h>
#include <stdint.h>

typedef float v2f __attribute__((ext_vector_type(2)));
typedef float v8f __attribute__((ext_vector_type(8)));

#define NUM_ENTITIES 100000
#define NUM_RELATIONS 500
#define DIM 200
#define BATCH 512

#define WMMA_F32_16x16x4(a, b, c) \
    __builtin_amdgcn_wmma_f32_16x16x4_f32(false, (a), false, (b), (short)0, (c), false, false)

// ---------------------------------------------------------------------------
// Kernel 1: gather + complex coefficient fusion.
// coef layout: [BATCH, 2*DIM]; cols [0,200) = coef_r, cols [200,400) = coef_i
// ---------------------------------------------------------------------------
__global__ __launch_bounds__(256) void complex_coef_kernel(
    const float* __restrict__ entity_r, const float* __restrict__ entity_i,
    const float* __restrict__ relation_r, const float* __restrict__ relation_i,
    const int* __restrict__ heads, const int* __restrict__ rels,
    float* __restrict__ coef) {
  int idx = blockIdx.x * blockDim.x + threadIdx.x;   // BATCH*DIM = 102400 exactly
  int b = idx / DIM;
  int d = idx - b * DIM;
  int h = heads[b];
  int r = rels[b];
  float sr = entity_r[(size_t)h * DIM + d];
  float si = entity_i[(size_t)h * DIM + d];
  float rr = relation_r[(size_t)r * DIM + d];
  float ri = relation_i[(size_t)r * DIM + d];
  coef[(size_t)b * (2 * DIM) + d]       = rr * sr - ri * si;  // coef_r
  coef[(size_t)b * (2 * DIM) + DIM + d] = rr * si + ri * sr;  // coef_i
}

// ---------------------------------------------------------------------------
// Kernel 2: f32 WMMA GEMM with async global->LDS staging of the B operand.
//   out[b, e] = sum_d coef_r[b,d]*ent_r[e,d] + coef_i[b,d]*ent_i[e,d]
//
// Block tile: 256(M) x 32(N), 8 waves stacked in M; wave tile 32x32
// (4 x v8f accumulators). Grid: (100000/32 = 3125) x (512/256 = 2). All
// dimensions divide exactly -> no masking, EXEC is all-ones at every WMMA.
//
// B staging: the 32 entity rows a block needs are CONTIGUOUS in memory
// (row-major [E, DIM], rows n0..n0+31 = 25600 bytes per table). They are
// copied once per block into LDS with GLOBAL_LOAD_ASYNC_TO_LDS_B128
// (ASYNCcnt-tracked, no VGPR staging), then every wave reads its B fragments
// via ds_load_b64. This removes the 8x redundant per-wave global loads of
// shared B data; only the 2 per-wave A-fragment loads stay global (coef is
// 0.8 MB, L2-resident).
//
// V_WMMA_F32_16X16X4_F32 fragment layouts (ISA 7.12.2, 32-bit):
//   A 16x4 : lane L -> row M=L%16, floats {K=2*(L/16), K=2*(L/16)+1}
//   B 4x16 : lane L -> col N=L%16, floats {K=2*(L/16), K=2*(L/16)+1}
//   C 16x16: VGPR r, lane L -> row r+8*(L/16), col L%16
// ---------------------------------------------------------------------------
__global__ __launch_bounds__(256) void complex_wmma_gemm_kernel(
    const float* __restrict__ coef, const float* __restrict__ ent_r,
    const float* __restrict__ ent_i, float* __restrict__ out) {
  __shared__ float lds_r[32 * DIM];   // 25600 B
  __shared__ float lds_i[32 * DIM];   // 25600 B  (51.2 KB total, << 320 KB/WGP)

  const int tid  = threadIdx.x;
  const int lane = tid & 31;
  const int wave = tid >> 5;
  const int l16  = lane & 15;
  const int half = lane >> 4;          // 0: K pair {0,1}, 1: K pair {2,3}
  const int koff = 2 * half;

  const int m0 = blockIdx.y * 256 + wave * 32;  // batch rows of this wave
  const int n0 = blockIdx.x * 32;               // entity cols of this block

  // ---- async copy: 25600 B per table = 1600 x 16B chunks.
  // 256 threads: 6 full rounds of 4096 B, then a 1024 B tail (64 threads). ----
  {
    const uint64_t g_r = (uint64_t)(uintptr_t)(ent_r + (size_t)n0 * DIM);
    const uint64_t g_i = (uint64_t)(uintptr_t)(ent_i + (size_t)n0 * DIM);
    // Flat->LDS hardware addressing takes addr[31:0] (ISA 10.2): low 32 bits
    // of the flat shared-pointer are the LDS byte address for async VDST.
    const uint32_t lr = (uint32_t)(uintptr_t)(void*)lds_r;
    const uint32_t li = (uint32_t)(uintptr_t)(void*)lds_i;
    const uint32_t off = (uint32_t)tid * 16u;
#pragma unroll
    for (int it = 0; it < 6; ++it) {
      uint32_t o  = off + (uint32_t)it * 4096u;
      uint32_t dr = lr + o;
      uint32_t di = li + o;
      asm volatile("global_load_async_to_lds_b128 %0, %1, %2"
                   :: "v"(dr), "v"(o), "s"(g_r) : "memory");
      asm volatile("global_load_async_to_lds_b128 %0, %1, %2"
                   :: "v"(di), "v"(o), "s"(g_i) : "memory");
    }
    if (tid < 64) {
      uint32_t o  = off + 6u * 4096u;
      uint32_t dr = lr + o;
      uint32_t di = li + o;
      asm volatile("global_load_async_to_lds_b128 %0, %1, %2"
                   :: "v"(dr), "v"(o), "s"(g_r) : "memory");
      asm volatile("global_load_async_to_lds_b128 %0, %1, %2"
                   :: "v"(di), "v"(o), "s"(g_i) : "memory");
    }
    // Wait for this wave's async transfers; barrier covers the other waves'.
    asm volatile("s_wait_asynccnt 0x0" ::: "memory");
  }
  __syncthreads();

  // A fragment base pointers (two 16-row A tiles), coef stride 2*DIM = 400
  const float* a0 = coef + (size_t)(m0 + l16) * (2 * DIM) + koff;
  const float* a1 = coef + (size_t)(m0 + 16 + l16) * (2 * DIM) + koff;
  // B fragment base pointers in LDS (two 16-col B tiles per table)
  const float* br0 = lds_r + (l16)      * DIM + koff;
  const float* br1 = lds_r + (l16 + 16) * DIM + koff;
  const float* bi0 = lds_i + (l16)      * DIM + koff;
  const float* bi1 = lds_i + (l16 + 16) * DIM + koff;

  v8f acc00 = {}, acc01 = {}, acc10 = {}, acc11 = {};

  // ---- real part: K = 0..199 over lds_r ----
#pragma unroll 5
  for (int k = 0; k < DIM; k += 4) {
    v2f fa0 = *(const v2f*)(a0 + k);
    v2f fa1 = *(const v2f*)(a1 + k);
    v2f fb0 = *(const v2f*)(br0 + k);
    v2f fb1 = *(const v2f*)(br1 + k);
    acc00 = WMMA_F32_16x16x4(fa0, fb0, acc00);
    acc01 = WMMA_F32_16x16x4(fa0, fb1, acc01);
    acc10 = WMMA_F32_16x16x4(fa1, fb0, acc10);
    acc11 = WMMA_F32_16x16x4(fa1, fb1, acc11);
  }

  // ---- imaginary part: K = 0..199 over lds_i (coef cols 200..399) ----
#pragma unroll 5
  for (int k = 0; k < DIM; k += 4) {
    v2f fa0 = *(const v2f*)(a0 + DIM + k);
    v2f fa1 = *(const v2f*)(a1 + DIM + k);
    v2f fb0 = *(const v2f*)(bi0 + k);
    v2f fb1 = *(const v2f*)(bi1 + k);
    acc00 = WMMA_F32_16x16x4(fa0, fb0, acc00);
    acc01 = WMMA_F32_16x16x4(fa0, fb1, acc01);
    acc10 = WMMA_F32_16x16x4(fa1, fb0, acc10);
    acc11 = WMMA_F32_16x16x4(fa1, fb1, acc11);
  }

  // ---- store: C/D layout (VGPR r, lane L) -> row r+8*half, col l16 ----
  const int col0 = n0 + l16;
#pragma unroll
  for (int r = 0; r < 8; ++r) {
    int row0 = m0 + r + 8 * half;       // A-tile 0 rows
    int row1 = row0 + 16;               // A-tile 1 rows
    out[(size_t)row0 * NUM_ENTITIES + col0]      = acc00[r];
    out[(size_t)row0 * NUM_ENTITIES + col0 + 16] = acc01[r];
    out[(size_t)row1 * NUM_ENTITIES + col0]      = acc10[r];
    out[(size_t)row1 * NUM_ENTITIES + col0 + 16] = acc11[r];
  }
}

extern "C" void kernel_launch(void* const* d_in, const int* in_sizes, int n_in,
                              void* d_out, int out_size, void* d_ws, size_t ws_size,
                              hipStream_t stream) {
  const float* entity_r   = (const float*)d_in[0];
  const float* entity_i   = (const float*)d_in[1];
  const float* relation_r = (const float*)d_in[2];
  const float* relation_i = (const float*)d_in[3];
  const int*   heads      = (const int*)d_in[4];
  const int*   rels       = (const int*)d_in[5];
  float*       out        = (float*)d_out;

  float* coef = (float*)d_ws;  // [BATCH, 2*DIM] = 512*400*4 = 819200 bytes

  // Kernel 1: BATCH*DIM = 102400 threads = 400 blocks of 256.
  complex_coef_kernel<<<dim3((BATCH * DIM) / 256), dim3(256), 0, stream>>>(
      entity_r, entity_i, relation_r, relation_i, heads, rels, coef);

  // Kernel 2: grid (3125, 2), block 256 (8 waves).
  complex_wmma_gemm_kernel<<<dim3(NUM_ENTITIES / 32, BATCH / 256), dim3(256), 0,
                             stream>>>(coef, entity_r, entity_i, out);
}